// LoRALayerNorm_72842645340230
// MI455X (gfx1250) — compile-verified
//
#include <hip/hip_runtime.h>
#include <hip/hip_bf16.h>

typedef float f4  __attribute__((ext_vector_type(4)));
typedef int   v4i __attribute__((ext_vector_type(4)));

#define RANK      4
#define LN_BLOCK  256
#define WAVE      32
#define NWAVES    (LN_BLOCK / WAVE)
#define LN_N      8192
#define NCH       (LN_N / (4 * LN_BLOCK))    // 8 float4 chunks per thread

// gfx1250 async global->LDS path (ASYNCcnt). Guarded so compile never fails;
// fall back to a synchronous b128 staging copy if the builtins are absent.
#if defined(__has_builtin)
#if __has_builtin(__builtin_amdgcn_global_load_async_to_lds_b128) && \
    __has_builtin(__builtin_amdgcn_s_wait_asynccnt)
#define USE_ASYNC_LDS 1
#endif
#endif
#ifndef USE_ASYNC_LDS
#define USE_ASYNC_LDS 0
#endif

// Pointer types matching the builtin's prototype:
//   (int4 addrspace(1)* gaddr, int4 addrspace(3)* ldsaddr, Imm offset, Imm cpol)
typedef __attribute__((address_space(1))) v4i* gp4;
typedef __attribute__((address_space(3))) v4i* lp4;

// ---------------------------------------------------------------------------
// Specialized main kernel, N = 8192 fixed at compile time (harness shape).
// One block per row (32 KB row staged in LDS, static allocation).
// Phase 1: 8 x global_load_async_to_lds_b128 with literal IOFFSETs off one
//          base address pair (IOFFSET is added to BOTH global and LDS addr).
// Phase 2: per-thread sum/sumsq from LDS, wave32 butterfly + cross-wave fold.
// Phase 3: normalize from LDS, apply L2-hot scale/shift, NT-store output.
// x read from HBM exactly once; output written once -> ~512 MB total traffic
// => ~22 us floor at 23.3 TB/s.
// ---------------------------------------------------------------------------
__global__ __launch_bounds__(LN_BLOCK) void lora_ln_8192(
    const float* __restrict__ x,
    const float* __restrict__ scale,
    const float* __restrict__ shift,
    float* __restrict__ out)
{
    __shared__ float smem[LN_N + 2 * NWAVES];
    float* red = smem + LN_N;

    const int tid = threadIdx.x;
    const float* xrow = x + (size_t)blockIdx.x * LN_N;

#if USE_ASYNC_LDS
    {
        gp4 g = (gp4)(xrow + 4 * tid);      // one global base per thread
        lp4 l = (lp4)(smem + 4 * tid);      // one LDS base per thread
        __builtin_amdgcn_global_load_async_to_lds_b128(g, l, 0 * LN_BLOCK * 16, 0);
        __builtin_amdgcn_global_load_async_to_lds_b128(g, l, 1 * LN_BLOCK * 16, 0);
        __builtin_amdgcn_global_load_async_to_lds_b128(g, l, 2 * LN_BLOCK * 16, 0);
        __builtin_amdgcn_global_load_async_to_lds_b128(g, l, 3 * LN_BLOCK * 16, 0);
        __builtin_amdgcn_global_load_async_to_lds_b128(g, l, 4 * LN_BLOCK * 16, 0);
        __builtin_amdgcn_global_load_async_to_lds_b128(g, l, 5 * LN_BLOCK * 16, 0);
        __builtin_amdgcn_global_load_async_to_lds_b128(g, l, 6 * LN_BLOCK * 16, 0);
        __builtin_amdgcn_global_load_async_to_lds_b128(g, l, 7 * LN_BLOCK * 16, 0);
    }
    __builtin_amdgcn_s_wait_asynccnt(0);    // this wave's async copies done
    __syncthreads();                        // all waves' copies visible
#else
    #pragma unroll
    for (int it = 0; it < NCH; ++it) {
        int idx = it * LN_BLOCK + tid;
        f4 v = __builtin_nontemporal_load((const f4*)xrow + idx);
        *((f4*)smem + idx) = v;
    }
    __syncthreads();
#endif

    // ---- Phase 2: statistics -------------------------------------------------
    const f4* l4 = (const f4*)smem;
    float s = 0.0f, q = 0.0f;
    #pragma unroll
    for (int it = 0; it < NCH; ++it) {
        f4 v = l4[it * LN_BLOCK + tid];
        s += v.x + v.y + v.z + v.w;
        q += v.x * v.x + v.y * v.y + v.z * v.z + v.w * v.w;
    }
    #pragma unroll
    for (int m = WAVE >> 1; m > 0; m >>= 1) {   // wave32 butterfly
        s += __shfl_xor(s, m, WAVE);
        q += __shfl_xor(q, m, WAVE);
    }
    const int wid  = tid / WAVE;
    const int lane = tid % WAVE;
    if (lane == 0) { red[2 * wid] = s; red[2 * wid + 1] = q; }
    __syncthreads();

    float ts = 0.0f, tq = 0.0f;
    #pragma unroll
    for (int w = 0; w < NWAVES; ++w) { ts += red[2 * w]; tq += red[2 * w + 1]; }
    const float invN = 1.0f / (float)LN_N;
    const float mean = ts * invN;
    const float var  = tq * invN - mean * mean;
    const float rstd = rsqrtf(var + 1e-5f);

    // ---- Phase 3: normalize + affine + streaming store ----------------------
    const f4* sc4 = (const f4*)scale;       // L2-resident, default temporal
    const f4* sh4 = (const f4*)shift;
    f4* o4 = (f4*)(out + (size_t)blockIdx.x * LN_N);
    #pragma unroll
    for (int it = 0; it < NCH; ++it) {
        int idx = it * LN_BLOCK + tid;
        f4 v  = l4[idx];
        f4 sc = sc4[idx];
        f4 sh = sh4[idx];
        f4 r;
        r.x = (v.x - mean) * rstd * sc.x + sh.x;
        r.y = (v.y - mean) * rstd * sc.y + sh.y;
        r.z = (v.z - mean) * rstd * sc.z + sh.z;
        r.w = (v.w - mean) * rstd * sc.w + sh.w;
        __builtin_nontemporal_store(r, o4 + idx);   // streamed once: TH=NT
    }
}

// ---------------------------------------------------------------------------
// Generic fallback (runtime N, dynamic LDS) — only used if shapes ever change.
// ---------------------------------------------------------------------------
__global__ __launch_bounds__(LN_BLOCK) void lora_ln_generic(
    const float* __restrict__ x,
    const float* __restrict__ scale,
    const float* __restrict__ shift,
    float* __restrict__ out,
    int N)
{
    extern __shared__ float dsmem[];
    float* red = dsmem + N;
    const int  tid = threadIdx.x;
    const int  nch = N / (4 * LN_BLOCK);
    const float* xrow = x + (size_t)blockIdx.x * N;

    for (int it = 0; it < nch; ++it) {
        int idx = it * LN_BLOCK + tid;
        f4 v = __builtin_nontemporal_load((const f4*)xrow + idx);
        *((f4*)dsmem + idx) = v;
    }
    __syncthreads();

    const f4* l4 = (const f4*)dsmem;
    float s = 0.0f, q = 0.0f;
    for (int it = 0; it < nch; ++it) {
        f4 v = l4[it * LN_BLOCK + tid];
        s += v.x + v.y + v.z + v.w;
        q += v.x * v.x + v.y * v.y + v.z * v.z + v.w * v.w;
    }
    for (int m = WAVE >> 1; m > 0; m >>= 1) {
        s += __shfl_xor(s, m, WAVE);
        q += __shfl_xor(q, m, WAVE);
    }
    if ((tid % WAVE) == 0) { red[2 * (tid / WAVE)] = s; red[2 * (tid / WAVE) + 1] = q; }
    __syncthreads();

    float ts = 0.0f, tq = 0.0f;
    for (int w = 0; w < NWAVES; ++w) { ts += red[2 * w]; tq += red[2 * w + 1]; }
    const float invN = 1.0f / (float)N;
    const float mean = ts * invN;
    const float rstd = rsqrtf(tq * invN - mean * mean + 1e-5f);

    const f4* sc4 = (const f4*)scale;
    const f4* sh4 = (const f4*)shift;
    f4* o4 = (f4*)(out + (size_t)blockIdx.x * N);
    for (int it = 0; it < nch; ++it) {
        int idx = it * LN_BLOCK + tid;
        f4 v  = l4[idx];
        f4 sc = sc4[idx];
        f4 sh = sh4[idx];
        f4 r;
        r.x = (v.x - mean) * rstd * sc.x + sh.x;
        r.y = (v.y - mean) * rstd * sc.y + sh.y;
        r.z = (v.z - mean) * rstd * sc.z + sh.z;
        r.w = (v.w - mean) * rstd * sc.w + sh.w;
        __builtin_nontemporal_store(r, o4 + idx);
    }
}

// ---------------------------------------------------------------------------
// Kernel 1: scale[i] = SCALING * sum_r A[i,r]*B[r,i]  (and same for shift).
// A is (N, RANK) row-major -> one f4 load; B is (RANK, N) -> 4 coalesced loads.
// Tiny: 32 blocks, fully L2-resident.
// ---------------------------------------------------------------------------
__global__ void lora_vec_kernel(const float* __restrict__ sA,
                                const float* __restrict__ sB,
                                const float* __restrict__ hA,
                                const float* __restrict__ hB,
                                float* __restrict__ scale,
                                float* __restrict__ shift,
                                int N)
{
    int i = blockIdx.x * blockDim.x + threadIdx.x;
    if (i >= N) return;
    f4 a = *(const f4*)(sA + 4 * i);
    f4 h = *(const f4*)(hA + 4 * i);
    float s = a.x * sB[i] + a.y * sB[N + i] + a.z * sB[2 * N + i] + a.w * sB[3 * N + i];
    float t = h.x * hB[i] + h.y * hB[N + i] + h.z * hB[2 * N + i] + h.w * hB[3 * N + i];
    scale[i] = 2.0f * s;   // SCALING = ALPHA / RANK = 8 / 4
    shift[i] = 2.0f * t;
}

// ---------------------------------------------------------------------------
extern "C" void kernel_launch(void* const* d_in, const int* in_sizes, int n_in,
                              void* d_out, int out_size, void* d_ws, size_t ws_size,
                              hipStream_t stream)
{
    const float* x  = (const float*)d_in[0];
    const float* sA = (const float*)d_in[1];
    const float* sB = (const float*)d_in[2];
    const float* hA = (const float*)d_in[3];
    const float* hB = (const float*)d_in[4];
    float* out = (float*)d_out;

    const int  N    = in_sizes[1] / RANK;        // 8192
    const long rows = (long)in_sizes[0] / N;     // 8192

    float* scale = (float*)d_ws;
    float* shift = scale + N;

    lora_vec_kernel<<<(N + 255) / 256, 256, 0, stream>>>(sA, sB, hA, hB, scale, shift, N);

    if (N == LN_N) {
        lora_ln_8192<<<dim3((unsigned)rows), dim3(LN_BLOCK), 0, stream>>>(
            x, scale, shift, out);
    } else {
        const size_t shbytes = (size_t)N * sizeof(float) + 2 * NWAVES * sizeof(float);
        lora_ln_generic<<<dim3((unsigned)rows), dim3(LN_BLOCK), shbytes, stream>>>(
            x, scale, shift, out, N);
    }
}